// simpleGAT_55207509623272
// MI455X (gfx1250) — compile-verified
//
#include <hip/hip_runtime.h>
#include <hip/hip_bf16.h>

typedef float v2f __attribute__((ext_vector_type(2)));
typedef float v8f __attribute__((ext_vector_type(8)));

#define NEG_SLOPE 0.2f
#define DH 128
#define HEADS 8
#define CH 16
#define NGRAPH 256

// ---------------------------------------------------------------------------
// fp32 WMMA: D = A(16x4) * B(4x16) + C(16x16), wave32.
// ---------------------------------------------------------------------------
__device__ __forceinline__ v8f wmma_f32_4(v2f a, v2f b, v8f c) {
  return __builtin_amdgcn_wmma_f32_16x16x4_f32(
      /*neg_a=*/false, a, /*neg_b=*/false, b,
      /*c_mod=*/(short)0, c, /*reuse_a=*/false, /*reuse_b=*/false);
}

// ---------------------------------------------------------------------------
// H[nrows,128] = X[nrows,FIN] @ W[FIN,128]   (FIN = 64 or 128)
// 256 threads = 8 waves; wave w computes rows [blk*128 + w*16, +16) x all 128
// cols. W staged in LDS in a fragment-native paired layout:
//   sW[kp][n] = {W[2kp][n], W[2kp+1][n]}  (float2)
// so each lane's B fragment is a single aligned ds_load_b64.
// EXEC is uniform: guards are clamps + masked stores (WMMA needs EXEC=all-1).
// ---------------------------------------------------------------------------
template <int FIN>
__global__ void __launch_bounds__(256) gemm_wmma_kernel(
    const float* __restrict__ X, const float* __restrict__ W,
    float* __restrict__ Hout, int nrows) {
  __shared__ v2f sW[(FIN / 2) * DH];
  // stage: pair W[2kp][n] and W[2kp+1][n] into one float2
  for (int i = threadIdx.x; i < (FIN / 2) * DH; i += 256) {
    int kp = i >> 7;          // i / DH
    int n = i & (DH - 1);     // i % DH
    v2f p;
    p.x = W[(2 * kp) * DH + n];
    p.y = W[(2 * kp + 1) * DH + n];
    sW[i] = p;
  }
  __syncthreads();

  const int wave = threadIdx.x >> 5;
  const int lane = threadIdx.x & 31;
  const int rowBase = blockIdx.x * 128 + wave * 16;
  const int m16 = lane & 15;
  const int hi = lane >> 4;        // 0 for lanes 0-15, 1 for lanes 16-31
  const int khalf = hi << 1;       // 0 or 2

  const v8f vzero = {};
  v8f acc[8];
#pragma unroll
  for (int t = 0; t < 8; ++t) acc[t] = vzero;

  int ar = rowBase + m16;
  if (ar > nrows - 1) ar = nrows - 1;  // clamp (keeps EXEC uniform)
  const float* xrow = X + (long)ar * FIN;

  for (int k = 0; k < FIN; k += 4) {
    // A fragment: lanes 0-15 -> K = k,k+1 ; lanes 16-31 -> K = k+2,k+3
    v2f a = *(const v2f*)(xrow + k + khalf);
    // B fragment base: pair index kp = k/2 + hi, column n = t*16 + m16
    const v2f* brow = sW + ((k >> 1) + hi) * DH + m16;
#pragma unroll
    for (int t = 0; t < 8; ++t) {
      v2f b = brow[t * 16];  // single ds_load_b64, no repacking
      acc[t] = wmma_f32_4(a, b, acc[t]);
    }
  }

  // C/D layout: VGPR v holds M = v (lanes 0-15) or v+8 (lanes 16-31), N = lane&15
  const int rOff = hi * 8;
#pragma unroll
  for (int t = 0; t < 8; ++t) {
#pragma unroll
    for (int v = 0; v < 8; ++v) {
      int r = rowBase + rOff + v;
      if (r < nrows) Hout[(long)r * DH + t * 16 + m16] = acc[t][v];
    }
  }
}

// ---------------------------------------------------------------------------
// alpha_src[n,h] = sum_c h[n,h,c]*a_src[h,c] ; same for dst. 1 thread/(n,h).
// ---------------------------------------------------------------------------
__global__ void __launch_bounds__(256) alpha_kernel(
    const float* __restrict__ Hh, const float* __restrict__ a_src,
    const float* __restrict__ a_dst, float* __restrict__ asrc,
    float* __restrict__ adst, int nNodes) {
  int t = blockIdx.x * blockDim.x + threadIdx.x;
  if (t >= nNodes * HEADS) return;
  int n = t >> 3, hd = t & 7;
  const float4* hv = (const float4*)(Hh + (long)n * DH + hd * CH);
  const float4* s4 = (const float4*)(a_src + hd * CH);
  const float4* d4 = (const float4*)(a_dst + hd * CH);
  float ss = 0.f, dd = 0.f;
#pragma unroll
  for (int i = 0; i < 4; ++i) {
    float4 h4 = hv[i], a = s4[i], b = d4[i];
    ss += h4.x * a.x + h4.y * a.y + h4.z * a.z + h4.w * a.w;
    dd += h4.x * b.x + h4.y * b.y + h4.z * b.z + h4.w * b.w;
  }
  asrc[t] = ss;
  adst[t] = dd;
}

// order-preserving float <-> uint map for atomicMax-based segment max
__device__ __forceinline__ unsigned fmap(float f) {
  unsigned u = __float_as_uint(f);
  return (u & 0x80000000u) ? ~u : (u | 0x80000000u);
}
__device__ __forceinline__ float funmap(unsigned u) {
  return __uint_as_float((u & 0x80000000u) ? (u & 0x7FFFFFFFu) : ~u);
}

__global__ void __launch_bounds__(256) edge_max_kernel(
    const int* __restrict__ src, const int* __restrict__ dst,
    const float* __restrict__ asrc, const float* __restrict__ adst,
    float* __restrict__ ebuf, unsigned* __restrict__ mbuf, int nEdges) {
  int e = blockIdx.x * blockDim.x + threadIdx.x;
  if (e >= nEdges) return;
  int s = src[e], d = dst[e];
#pragma unroll
  for (int h = 0; h < HEADS; ++h) {
    float v = asrc[s * HEADS + h] + adst[d * HEADS + h];
    v = v > 0.f ? v : NEG_SLOPE * v;  // leaky relu
    ebuf[(long)e * HEADS + h] = v;
    atomicMax(&mbuf[d * HEADS + h], fmap(v));
  }
}

__global__ void __launch_bounds__(256) edge_sum_kernel(
    const int* __restrict__ dst, float* __restrict__ ebuf,
    const unsigned* __restrict__ mbuf, float* __restrict__ denom, int nEdges) {
  int e = blockIdx.x * blockDim.x + threadIdx.x;
  if (e >= nEdges) return;
  int d = dst[e];
#pragma unroll
  for (int h = 0; h < HEADS; ++h) {
    float m = funmap(mbuf[d * HEADS + h]);
    float ex = __expf(ebuf[(long)e * HEADS + h] - m);
    ebuf[(long)e * HEADS + h] = ex;  // in-place: e -> exp(e - m)
    atomicAdd(&denom[d * HEADS + h], ex);
  }
}

// one wave per edge; lane handles 4 channels (head = lane>>2)
__global__ void __launch_bounds__(256) edge_agg_kernel(
    const int* __restrict__ src, const int* __restrict__ dst,
    const float* __restrict__ ebuf, const float* __restrict__ denom,
    const float* __restrict__ Hh, float* __restrict__ Out, int nEdges) {
  long t = (long)blockIdx.x * blockDim.x + threadIdx.x;
  int e = (int)(t >> 5);
  if (e >= nEdges) return;
  int lane = (int)(t & 31);
  int s = src[e], d = dst[e];
  // hide the random-gather latency of the next edge's source row
  int enext = e + (int)(gridDim.x * blockDim.x >> 5);
  if (enext < nEdges)
    __builtin_prefetch(Hh + (long)src[enext] * DH + lane * 4, 0, 3);
  int hd = lane >> 2;
  float alpha = ebuf[(long)e * HEADS + hd] / denom[d * HEADS + hd];
  float4 hv = *(const float4*)(Hh + (long)s * DH + lane * 4);
  float* o = Out + (long)d * DH + lane * 4;
  atomicAdd(o + 0, alpha * hv.x);
  atomicAdd(o + 1, alpha * hv.y);
  atomicAdd(o + 2, alpha * hv.z);
  atomicAdd(o + 3, alpha * hv.w);
}

__global__ void __launch_bounds__(256) bias_relu_kernel(
    float* __restrict__ Out, const float* __restrict__ b, int nNodes) {
  int t = blockIdx.x * blockDim.x + threadIdx.x;
  if (t >= nNodes * DH) return;
  float v = Out[t] + b[t & (DH - 1)];
  Out[t] = v > 0.f ? v : 0.f;
}

// one wave per node; lane handles 4 channels
__global__ void __launch_bounds__(256) pool_kernel(
    const float* __restrict__ Hh, const int* __restrict__ batch,
    float* __restrict__ pooled, float* __restrict__ counts, int nNodes) {
  long t = (long)blockIdx.x * blockDim.x + threadIdx.x;
  int n = (int)(t >> 5);
  if (n >= nNodes) return;
  int lane = (int)(t & 31);
  int g = batch[n];
  float4 hv = *(const float4*)(Hh + (long)n * DH + lane * 4);
  float* p = pooled + (long)g * DH + lane * 4;
  atomicAdd(p + 0, hv.x);
  atomicAdd(p + 1, hv.y);
  atomicAdd(p + 2, hv.z);
  atomicAdd(p + 3, hv.w);
  if (lane == 0) atomicAdd(&counts[g], 1.0f);
}

__global__ void __launch_bounds__(256) classifier_kernel(
    const float* __restrict__ pooled, const float* __restrict__ counts,
    const float* __restrict__ Wc1, const float* __restrict__ bc1,
    const float* __restrict__ Wc2, const float* __restrict__ bc2,
    const float* __restrict__ Wc3, const float* __restrict__ bc3,
    float* __restrict__ out) {
  int g = blockIdx.x * blockDim.x + threadIdx.x;
  if (g >= NGRAPH) return;
  float inv = 1.0f / fmaxf(counts[g], 1.0f);
  float z1[16];
#pragma unroll 4
  for (int j = 0; j < 16; ++j) {
    float acc = bc1[j];
    for (int i = 0; i < DH; ++i) acc += pooled[g * DH + i] * inv * Wc1[i * 16 + j];
    z1[j] = fmaxf(acc, 0.f);
  }
  float z2[8];
#pragma unroll
  for (int j = 0; j < 8; ++j) {
    float acc = bc2[j];
    for (int i = 0; i < 16; ++i) acc += z1[i] * Wc2[i * 8 + j];
    z2[j] = fmaxf(acc, 0.f);
  }
#pragma unroll
  for (int j = 0; j < 2; ++j) {
    float acc = bc3[j];
    for (int i = 0; i < 8; ++i) acc += z2[i] * Wc3[i * 2 + j];
    out[g * 2 + j] = acc;
  }
}

// ---------------------------------------------------------------------------
extern "C" void kernel_launch(void* const* d_in, const int* in_sizes, int n_in,
                              void* d_out, int out_size, void* d_ws,
                              size_t ws_size, hipStream_t stream) {
  (void)n_in; (void)out_size; (void)ws_size;

  const float* x = (const float*)d_in[0];
  const int* ei = (const int*)d_in[1];
  const int* batch = (const int*)d_in[2];
  const int nEdges = in_sizes[1] / 2;   // 850000 (E + N self loops)
  const int nNodes = in_sizes[2];       // 50000
  const int* src = ei;                  // edge_index[0]
  const int* dst = ei + nEdges;         // edge_index[1]

  const float* Wl[3] = {(const float*)d_in[3], (const float*)d_in[7],
                        (const float*)d_in[11]};
  const float* ASl[3] = {(const float*)d_in[4], (const float*)d_in[8],
                         (const float*)d_in[12]};
  const float* ADl[3] = {(const float*)d_in[5], (const float*)d_in[9],
                         (const float*)d_in[13]};
  const float* Bl[3] = {(const float*)d_in[6], (const float*)d_in[10],
                        (const float*)d_in[14]};
  const float* Wc1 = (const float*)d_in[15];
  const float* bc1 = (const float*)d_in[16];
  const float* Wc2 = (const float*)d_in[17];
  const float* bc2 = (const float*)d_in[18];
  const float* Wc3 = (const float*)d_in[19];
  const float* bc3 = (const float*)d_in[20];

  // workspace layout (floats)
  float* ws = (float*)d_ws;
  size_t off = 0;
  float* hA = ws + off;     off += (size_t)nNodes * DH;   // gemm result
  float* hB = ws + off;     off += (size_t)nNodes * DH;   // layer output
  float* asrc = ws + off;   off += (size_t)nNodes * HEADS;
  float* adst = ws + off;   off += (size_t)nNodes * HEADS;
  unsigned* mbuf = (unsigned*)(ws + off); off += (size_t)nNodes * HEADS;
  float* denom = ws + off;  off += (size_t)nNodes * HEADS;
  float* ebuf = ws + off;   off += (size_t)nEdges * HEADS;
  float* pooled = ws + off; off += (size_t)NGRAPH * DH;
  float* counts = ws + off; off += (size_t)NGRAPH;

  auto cdiv = [](long a, long b) { return (int)((a + b - 1) / b); };

  const int gemmBlocks = cdiv(nNodes, 128);
  const int nhBlocks = cdiv((long)nNodes * HEADS, 256);
  const int edgeBlocks = cdiv(nEdges, 256);
  const int aggBlocks = cdiv((long)nEdges * 32, 256);
  const int ncBlocks = cdiv((long)nNodes * DH, 256);
  const int poolBlocks = cdiv((long)nNodes * 32, 256);

  const float* lin = x;
  for (int l = 0; l < 3; ++l) {
    // 1) h = lin @ W   (fp32 WMMA)
    if (l == 0)
      gemm_wmma_kernel<64><<<gemmBlocks, 256, 0, stream>>>(lin, Wl[l], hA, nNodes);
    else
      gemm_wmma_kernel<128><<<gemmBlocks, 256, 0, stream>>>(lin, Wl[l], hA, nNodes);
    // 2) per-node attention logits
    alpha_kernel<<<nhBlocks, 256, 0, stream>>>(hA, ASl[l], ADl[l], asrc, adst, nNodes);
    // 3) zero accumulators (after gemm consumed hB)
    hipMemsetAsync(mbuf, 0, (size_t)nNodes * HEADS * 4, stream);
    hipMemsetAsync(denom, 0, (size_t)nNodes * HEADS * 4, stream);
    hipMemsetAsync(hB, 0, (size_t)nNodes * DH * 4, stream);
    // 4) segment softmax + aggregation
    edge_max_kernel<<<edgeBlocks, 256, 0, stream>>>(src, dst, asrc, adst, ebuf, mbuf, nEdges);
    edge_sum_kernel<<<edgeBlocks, 256, 0, stream>>>(dst, ebuf, mbuf, denom, nEdges);
    edge_agg_kernel<<<aggBlocks, 256, 0, stream>>>(src, dst, ebuf, denom, hA, hB, nEdges);
    // 5) + bias, relu
    bias_relu_kernel<<<ncBlocks, 256, 0, stream>>>(hB, Bl[l], nNodes);
    lin = hB;
  }

  // global mean pool + classifier
  hipMemsetAsync(pooled, 0, (size_t)NGRAPH * DH * 4, stream);
  hipMemsetAsync(counts, 0, (size_t)NGRAPH * 4, stream);
  pool_kernel<<<poolBlocks, 256, 0, stream>>>(hB, batch, pooled, counts, nNodes);
  classifier_kernel<<<1, 256, 0, stream>>>(pooled, counts, Wc1, bc1, Wc2, bc2,
                                           Wc3, bc3, (float*)d_out);
}